// gcn_27805618274407
// MI455X (gfx1250) — compile-verified
//
#include <hip/hip_runtime.h>
#include <hip/hip_bf16.h>
#include <math.h>

#define N_NODES 50000
#define N_EDGES 800000
#define DIM 64

typedef float v2f __attribute__((ext_vector_type(2)));
typedef float v8f __attribute__((ext_vector_type(8)));

// ---------------- degree accumulation ----------------
__global__ void __launch_bounds__(256)
gcn_deg_kernel(const int* __restrict__ src, const int* __restrict__ dst,
               float* __restrict__ no, float* __restrict__ ni, int nE) {
    int e = blockIdx.x * blockDim.x + threadIdx.x;
    if (e >= nE) return;
    atomicAdd(no + src[e], 1.0f);
    atomicAdd(ni + dst[e], 1.0f);
}

// deg -> rsqrt(max(deg,1)) in place
__global__ void __launch_bounds__(256)
gcn_norm_kernel(float* __restrict__ no, float* __restrict__ ni, int n) {
    int i = blockIdx.x * blockDim.x + threadIdx.x;
    if (i >= n) return;
    no[i] = rsqrtf(fmaxf(no[i], 1.0f));
    ni[i] = rsqrtf(fmaxf(ni[i], 1.0f));
}

// agg[dst[e], j] += vals[src[e], j] * (scale ? scale[src[e]] : 1)
// Thread per (edge, feature): gather + atomic scatter both fully coalesced
// (64 consecutive lanes cover one row; index loads are wave-uniform broadcasts).
__global__ void __launch_bounds__(256)
gcn_scatter_kernel(const int* __restrict__ src, const int* __restrict__ dst,
                   const float* __restrict__ vals, const float* __restrict__ scale,
                   float* __restrict__ agg, long long total) {
    long long idx = (long long)blockIdx.x * blockDim.x + threadIdx.x;
    if (idx >= total) return;
    int e = (int)(idx >> 6);
    int j = (int)(idx & 63);
    int s = src[e], d = dst[e];
    float v = vals[(size_t)s * DIM + j];
    if (scale) v *= scale[s];                    // layer-1: fold out_norm row-scale
    atomicAdd(agg + (size_t)d * DIM + j, v);
}

// hs2 = relu((agg * in_norm[:,None]) @ W1 + b1) * out_norm[:,None]
// One block = 16 rows; 4 waves each own one 16-wide N-tile; K=64 via 16 WMMA f32 16x16x4.
__global__ void __launch_bounds__(128)
gcn_layer1_wmma(const float* __restrict__ agg, const float* __restrict__ W1,
                const float* __restrict__ b1, const float* __restrict__ ni,
                const float* __restrict__ no, float* __restrict__ hs2) {
    __shared__ float As[16][68];                 // pitch 68: conflict-free column reads
    const int tid    = threadIdx.x;
    const int wave   = tid >> 5;
    const int lane   = tid & 31;
    const int half   = lane >> 4;                // 0: lanes 0-15, 1: lanes 16-31
    const int lanelo = lane & 15;
    const int m0     = blockIdx.x * 16;          // 50000 % 16 == 0 -> always full tile

    // Stage A tile, pre-scaled by in_norm (row scale of agg)
    for (int t = tid; t < 16 * DIM; t += 128) {
        int r = t >> 6, c = t & 63;
        As[r][c] = agg[(size_t)(m0 + r) * DIM + c] * ni[m0 + r];
    }
    __syncthreads();

    const int n0 = wave * 16;
    v8f acc = {};
    #pragma unroll
    for (int k0 = 0; k0 < DIM; k0 += 4) {
        const int ka = k0 + 2 * half;
        v2f a, b;
        // A 16x4 f32 layout: lane half 0 -> K = k0,k0+1 ; half 1 -> K = k0+2,k0+3
        a.x = As[lanelo][ka];
        a.y = As[lanelo][ka + 1];
        // B 4x16 f32 layout: same K split across lane halves, N = n0+lanelo per lane
        b.x = W1[(size_t)ka * DIM + n0 + lanelo];
        b.y = W1[(size_t)(ka + 1) * DIM + n0 + lanelo];
        acc = __builtin_amdgcn_wmma_f32_16x16x4_f32(
            /*neg_a=*/false, a, /*neg_b=*/false, b,
            /*c_mod=*/(short)0, acc, /*reuse_a=*/false, /*reuse_b=*/false);
    }

    const int col  = n0 + lanelo;
    const float bias = b1[col];
    #pragma unroll
    for (int v = 0; v < 8; ++v) {                // C/D: VGPR v -> rows v / v+8
        int row = m0 + v + 8 * half;
        float x = acc[v] + bias;
        x = x > 0.0f ? x : 0.0f;                 // ReLU
        hs2[(size_t)row * DIM + col] = x * no[row];  // fold layer-2 out_norm scale
    }
}

// out[i] = sigmoid(in_norm[i] * dot(agg2[i,:], W2) + b2)  — one wave32 per node
__global__ void __launch_bounds__(256)
gcn_layer2_out(const float* __restrict__ agg2, const float* __restrict__ W2,
               const float* __restrict__ b2, const float* __restrict__ ni,
               float* __restrict__ out, int n) {
    const int wave = threadIdx.x >> 5;
    const int lane = threadIdx.x & 31;
    const int node = blockIdx.x * 8 + wave;
    if (node >= n) return;
    const float* row = agg2 + (size_t)node * DIM;
    float s = row[lane] * W2[lane] + row[lane + 32] * W2[lane + 32];
    #pragma unroll
    for (int off = 16; off > 0; off >>= 1)
        s += __shfl_xor(s, off, 32);             // wave32 butterfly reduction
    if (lane == 0) {
        float x = ni[node] * s + b2[0];
        out[node] = 1.0f / (1.0f + expf(-x));
    }
}

extern "C" void kernel_launch(void* const* d_in, const int* in_sizes, int n_in,
                              void* d_out, int out_size, void* d_ws, size_t ws_size,
                              hipStream_t stream) {
    const float* feat = (const float*)d_in[0];
    const float* W1   = (const float*)d_in[1];
    const float* b1   = (const float*)d_in[2];
    const float* W2   = (const float*)d_in[3];
    const float* b2   = (const float*)d_in[4];
    const int*   src  = (const int*)d_in[5];
    const int*   dst  = (const int*)d_in[6];
    float* out = (float*)d_out;

    // Workspace layout (256B-aligned offsets), ~26 MB total, resident in L2
    const size_t NODE_B = 50048 * sizeof(float);                 // padded node vec
    const size_t ROW_B  = (size_t)N_NODES * DIM * sizeof(float); // 12.8 MB
    char* ws = (char*)d_ws;
    float* p_no = (float*)(ws);                           // out_norm
    float* p_ni = (float*)(ws + NODE_B);                  // in_norm
    float* bufB = (float*)(ws + 2 * NODE_B);              // agg1, then agg2
    float* bufC = (float*)(ws + 2 * NODE_B + ROW_B);      // hs2

    const long long totEF = (long long)N_EDGES * DIM;     // 51.2M (edge,feat) threads
    const int scatterGrid = (int)((totEF + 255) / 256);

    // zero degree accumulators + agg1
    hipMemsetAsync(p_no, 0, 2 * NODE_B, stream);
    hipMemsetAsync(bufB, 0, ROW_B, stream);

    // degrees -> norms
    gcn_deg_kernel<<<(N_EDGES + 255) / 256, 256, 0, stream>>>(src, dst, p_no, p_ni, N_EDGES);
    gcn_norm_kernel<<<(N_NODES + 255) / 256, 256, 0, stream>>>(p_no, p_ni, N_NODES);

    // layer 1: fused (out_norm-scale + gather + scatter-add), then WMMA GEMM epilogue
    gcn_scatter_kernel<<<scatterGrid, 256, 0, stream>>>(src, dst, feat, p_no, bufB, totEF);
    gcn_layer1_wmma<<<N_NODES / 16, 128, 0, stream>>>(bufB, W1, b1, p_ni, p_no, bufC);

    // layer 2: reuse bufB as agg2 (agg1 fully consumed above)
    hipMemsetAsync(bufB, 0, ROW_B, stream);
    gcn_scatter_kernel<<<scatterGrid, 256, 0, stream>>>(src, dst, bufC, nullptr, bufB, totEF);
    gcn_layer2_out<<<(N_NODES + 7) / 8, 256, 0, stream>>>(bufB, W2, b2, p_ni, out, N_NODES);
}